// AttentionGCN_90907277787612
// MI455X (gfx1250) — compile-verified
//
#include <hip/hip_runtime.h>

#define N_NODES 100000
#define N_EDGES 1600000
#define DIM 64

typedef float v2f __attribute__((ext_vector_type(2)));
typedef float v8f __attribute__((ext_vector_type(8)));

// ---------------------------------------------------------------------------
// K1: per-node attention scores s_row = x.w_r, s_col = x.w_c ; deg init = 1.0
// ---------------------------------------------------------------------------
__global__ __launch_bounds__(256) void k_scores(const float* __restrict__ x,
                                                const float* __restrict__ lin_w,
                                                float* __restrict__ s_row,
                                                float* __restrict__ s_col,
                                                float* __restrict__ deg) {
    int i = blockIdx.x * blockDim.x + threadIdx.x;
    if (i >= N_NODES) return;
    const float4* xr = (const float4*)(x + (size_t)i * DIM);
    const float4* wr = (const float4*)(lin_w);
    const float4* wc = (const float4*)(lin_w + DIM);
    float sr = 0.f, sc = 0.f;
#pragma unroll
    for (int j = 0; j < DIM / 4; ++j) {
        float4 v = xr[j], a = wr[j], b = wc[j];
        sr += v.x * a.x + v.y * a.y + v.z * a.z + v.w * a.w;
        sc += v.x * b.x + v.y * b.y + v.z * b.z + v.w * b.w;
    }
    s_row[i] = sr;
    s_col[i] = sc;
    deg[i] = 1.0f;  // self-loop weight
}

// ---------------------------------------------------------------------------
// K2: xw = x @ conv_w.T via V_WMMA_F32_16X16X4_F32.
// One wave per 16-row tile. A (16x4 f32): lane<16 holds {K,K+1} of row M=lane,
// lane>=16 holds {K+2,K+3} of row M=lane-16 (ISA 7.12.2). B (4x16) mirrors A
// with N=lane&15. C/D: VGPR i -> M = i + 8*(lane>=16), N = lane&15.
// ---------------------------------------------------------------------------
__global__ __launch_bounds__(256) void k_xw_wmma(const float* __restrict__ x,
                                                 const float* __restrict__ convw,
                                                 float* __restrict__ xw) {
    const int lane = threadIdx.x & 31;
    const int wave = threadIdx.x >> 5;
    const int tile = blockIdx.x * 8 + wave;
    if (tile >= N_NODES / 16) return;  // wave-uniform exit (EXEC stays full)

    const int row0 = tile * 16;
    const int halfsel = lane >> 4;   // selects K pair {0,1} vs {2,3}
    const int lid = lane & 15;       // M for A, N for B

    // A: x[row0+lid][4*kk + 2*halfsel .. +1], kk = 0..15 (K = 64)
    v2f a[16];
    const float* arow = x + (size_t)(row0 + lid) * DIM + halfsel * 2;
#pragma unroll
    for (int kk = 0; kk < 16; ++kk) a[kk] = *(const v2f*)(arow + kk * 4);

#pragma unroll
    for (int nt = 0; nt < 4; ++nt) {
        const int n0 = nt * 16;
        v8f c = {0.f, 0.f, 0.f, 0.f, 0.f, 0.f, 0.f, 0.f};
        // B[k][n] = conv_w[n][k] -> contiguous in k per lane
        const float* brow = convw + (size_t)(n0 + lid) * DIM + halfsel * 2;
#pragma unroll
        for (int kk = 0; kk < 16; ++kk) {
            v2f b = *(const v2f*)(brow + kk * 4);
            c = __builtin_amdgcn_wmma_f32_16x16x4_f32(
                /*neg_a=*/false, a[kk], /*neg_b=*/false, b,
                /*c_mod=*/(short)0, c, /*reuse_a=*/false, /*reuse_b=*/false);
        }
        float* orow = xw + (size_t)(row0 + halfsel * 8) * DIM + n0 + lid;
#pragma unroll
        for (int i = 0; i < 8; ++i) orow[(size_t)i * DIM] = c[i];
    }
}

// ---------------------------------------------------------------------------
// K3: attn[e] = sigmoid(s_row[row]+s_col[col]+b) * ew[e]; deg[col] += attn
// ---------------------------------------------------------------------------
__global__ __launch_bounds__(256) void k_edge_attn(const long long* __restrict__ ei,
                                                   const float* __restrict__ ew,
                                                   const float* __restrict__ s_row,
                                                   const float* __restrict__ s_col,
                                                   const float* __restrict__ lin_b,
                                                   float* __restrict__ attn,
                                                   float* __restrict__ deg) {
    int e = blockIdx.x * blockDim.x + threadIdx.x;
    if (e >= N_EDGES) return;
    long long r = ei[e];
    long long c = ei[N_EDGES + e];
    float s = s_row[r] + s_col[c] + lin_b[0];
    float a = ew[e] / (1.0f + __expf(-s));
    attn[e] = a;
    atomicAdd(&deg[c], a);
}

// ---------------------------------------------------------------------------
// K4: deg -> deg_inv_sqrt (in place)
// ---------------------------------------------------------------------------
__global__ __launch_bounds__(256) void k_dinv(float* __restrict__ deg) {
    int i = blockIdx.x * blockDim.x + threadIdx.x;
    if (i >= N_NODES) return;
    float d = deg[i];
    deg[i] = (d > 0.f) ? rsqrtf(d) : 0.f;
}

// ---------------------------------------------------------------------------
// K5: out = conv_b + dinv^2 * xw   (bias + self-loop contribution)
// ---------------------------------------------------------------------------
__global__ __launch_bounds__(256) void k_out_init(const float* __restrict__ xw,
                                                  const float* __restrict__ dinv,
                                                  const float* __restrict__ conv_b,
                                                  float* __restrict__ out) {
    size_t i = (size_t)blockIdx.x * blockDim.x + threadIdx.x;
    if (i >= (size_t)N_NODES * DIM) return;
    int node = (int)(i >> 6);
    int j = (int)(i & 63);
    float di = dinv[node];
    out[i] = conv_b[j] + di * di * xw[i];
}

// ---------------------------------------------------------------------------
// K6: out[col] += xw[row] * (dinv[row]*attn*dinv[col]); 16 lanes/edge, float4
// ---------------------------------------------------------------------------
__global__ __launch_bounds__(256) void k_scatter(const long long* __restrict__ ei,
                                                 const float* __restrict__ attn,
                                                 const float* __restrict__ dinv,
                                                 const float* __restrict__ xw,
                                                 float* __restrict__ out) {
    size_t t = (size_t)blockIdx.x * blockDim.x + threadIdx.x;
    size_t e = t >> 4;
    if (e >= N_EDGES) return;
    int fo = (int)(t & 15) * 4;
    long long r = ei[e];
    long long c = ei[N_EDGES + e];
    float norm = dinv[r] * attn[e] * dinv[c];
    float4 v = *(const float4*)(xw + (size_t)r * DIM + fo);
    float* dst = out + (size_t)c * DIM + fo;
    atomicAdd(dst + 0, v.x * norm);
    atomicAdd(dst + 1, v.y * norm);
    atomicAdd(dst + 2, v.z * norm);
    atomicAdd(dst + 3, v.w * norm);
}

extern "C" void kernel_launch(void* const* d_in, const int* in_sizes, int n_in,
                              void* d_out, int out_size, void* d_ws, size_t ws_size,
                              hipStream_t stream) {
    const float* x      = (const float*)d_in[0];      // [N,64]
    const long long* ei = (const long long*)d_in[1];  // [2,E] int64
    const float* ew     = (const float*)d_in[2];      // [E]
    const float* lin_w  = (const float*)d_in[3];      // [1,128]
    const float* lin_b  = (const float*)d_in[4];      // [1]
    const float* conv_w = (const float*)d_in[5];      // [64,64]
    const float* conv_b = (const float*)d_in[6];      // [64]
    float* out = (float*)d_out;

    // Workspace partition (16B-aligned slabs)
    char* ws = (char*)d_ws;
    float* s_row = (float*)ws;                                  // N floats
    float* s_col = s_row + N_NODES;                             // N floats
    float* deg   = s_col + N_NODES;                             // N floats (-> dinv)
    float* xw    = deg + N_NODES;                               // N*64 floats
    float* attn  = xw + (size_t)N_NODES * DIM;                  // E floats

    // K1: scores + deg init
    k_scores<<<(N_NODES + 255) / 256, 256, 0, stream>>>(x, lin_w, s_row, s_col, deg);

    // K2: WMMA GEMM (6250 tiles, 8 waves/block)
    k_xw_wmma<<<(N_NODES / 16 + 7) / 8, 256, 0, stream>>>(x, conv_w, xw);

    // K3: edge attention + degree accumulation
    k_edge_attn<<<(N_EDGES + 255) / 256, 256, 0, stream>>>(ei, ew, s_row, s_col, lin_b,
                                                           attn, deg);

    // K4: deg -> 1/sqrt(deg)
    k_dinv<<<(N_NODES + 255) / 256, 256, 0, stream>>>(deg);

    // K5: out = conv_b + dinv^2 * xw
    k_out_init<<<((size_t)N_NODES * DIM + 255) / 256, 256, 0, stream>>>(xw, deg, conv_b, out);

    // K6: edge scatter (16 lanes per edge)
    k_scatter<<<((size_t)N_EDGES * 16 + 255) / 256, 256, 0, stream>>>(ei, attn, deg, xw, out);
}